// RegionLoss_17068200034567
// MI455X (gfx1250) — compile-verified
//
#include <hip/hip_runtime.h>
#include <math.h>

#define NB    32
#define NA    5
#define NC    80
#define NH    38
#define NW    38
#define NHW   (NH * NW)      /* 1444 */
#define NHW4  (NHW / 4)      /* 361  */
#define CELLS (NA * NHW)     /* 7220 */
#define CHUNKS (CELLS / 4)   /* 1805 */
#define MAXT  50
#define SIL   0.6f

typedef __attribute__((ext_vector_type(2))) float v2f;
typedef __attribute__((ext_vector_type(4))) float v4f;
typedef __attribute__((ext_vector_type(8))) float v8f;

__constant__ float d_anchors[2 * NA] = {
    1.3221f, 1.73145f, 3.19275f, 4.00944f, 5.05587f,
    8.09892f, 9.47112f, 4.84053f, 11.2364f, 10.0071f};

__global__ void zero_kernel(float* o, int n) {
  int i = blockIdx.x * blockDim.x + threadIdx.x;
  if (i < n) o[i] = 0.0f;
}

__device__ __forceinline__ float fsigmoid(float v) {
  return 1.0f / (1.0f + __expf(-v));
}

__global__ __launch_bounds__(256) void region_loss_kernel(
    const float* __restrict__ out, const float* __restrict__ tgt,
    float* __restrict__ loss_out) {
  __shared__ float s_raw[MAXT * 5];
  __shared__ float s_ex0[MAXT], s_ex1[MAXT], s_ey0[MAXT], s_ey1[MAXT];
  __shared__ float s_gw[MAXT], s_gh[MAXT], s_area[MAXT];
  __shared__ float s_txv[MAXT], s_tyv[MAXT], s_twv[MAXT], s_thv[MAXT];
  __shared__ int s_bn[MAXT], s_gi[MAXT], s_gj[MAXT], s_cls[MAXT];
  __shared__ int s_valid[MAXT], s_win[MAXT];
  __shared__ float s_wsum[8];

  const int tid = threadIdx.x;
  const int b = blockIdx.y;

  // ---- Stage target[b] (250 f32) into LDS via CDNA5 async global->LDS DMA ----
  if (tid < MAXT * 5) {
    unsigned lds_off = (unsigned)(size_t)&s_raw[tid];
    unsigned long long ga =
        (unsigned long long)(const void*)(tgt + (size_t)b * (MAXT * 5) + tid);
    asm volatile("global_load_async_to_lds_b32 %0, %1, off"
                 :: "v"(lds_off), "v"(ga)
                 : "memory");
  }
#if __has_builtin(__builtin_amdgcn_s_wait_asynccnt)
  __builtin_amdgcn_s_wait_asynccnt(0);
#else
  asm volatile("s_wait_asynccnt 0x0" ::: "memory");
#endif
  __syncthreads();

  // ---- Per-target prep (parallel over t) ----
  if (tid < MAXT) {
    const float* r = &s_raw[tid * 5];
    float gx = r[1] * NW, gy = r[2] * NH, gw = r[3] * NW, gh = r[4] * NH;
    s_gw[tid] = gw;
    s_gh[tid] = gh;
    s_ex0[tid] = gx - gw * 0.5f;
    s_ex1[tid] = gx + gw * 0.5f;
    s_ey0[tid] = gy - gh * 0.5f;
    s_ey1[tid] = gy + gh * 0.5f;
    s_area[tid] = gw * gh;
    // best anchor: concentric-box IoU = min(gw,aw)*min(gh,ah) / (areas - inter)
    int bn = 0;
    float best = -1.0f;
    for (int a = 0; a < NA; a++) {
      float aw = d_anchors[2 * a], ah = d_anchors[2 * a + 1];
      float inter = fminf(gw, aw) * fminf(gh, ah);
      float uni = gw * gh + aw * ah - inter;
      float iou = inter / fmaxf(uni, 1e-12f);
      if (iou > best) { best = iou; bn = a; }   // first-max like jnp.argmax
    }
    int gi = (int)floorf(gx); gi = min(max(gi, 0), NW - 1);
    int gj = (int)floorf(gy); gj = min(max(gj, 0), NH - 1);
    s_bn[tid] = bn;
    s_gi[tid] = gi;
    s_gj[tid] = gj;
    s_txv[tid] = gx - (float)gi;
    s_tyv[tid] = gy - (float)gj;
    s_twv[tid] = __logf(fmaxf(gw, 1e-12f) / d_anchors[2 * bn]);
    s_thv[tid] = __logf(fmaxf(gh, 1e-12f) / d_anchors[2 * bn + 1]);
    int cv = (int)r[0];
    s_cls[tid] = min(max(cv, 0), NC - 1);
    s_valid[tid] = (r[1] != 0.0f) ? 1 : 0;
  }
  __syncthreads();
  if (tid == 0) {  // valid = cumprod(t.x != 0)
    int run = 1;
    for (int t = 0; t < MAXT; t++) { run &= s_valid[t]; s_valid[t] = run; }
  }
  __syncthreads();
  if (tid < MAXT) {  // winner = valid and no later valid target hits same slot
    int w = s_valid[tid];
    if (w) {
      for (int t2 = tid + 1; t2 < MAXT; t2++) {
        if (s_valid[t2] && s_bn[t2] == s_bn[tid] && s_gi[t2] == s_gi[tid] &&
            s_gj[t2] == s_gj[tid]) { w = 0; break; }
      }
    }
    s_win[tid] = w;
  }
  __syncthreads();

  // ---- Main streaming pass: one thread per 4 consecutive cells ----
  // 1444 % 4 == 0 -> a 4-cell chunk never crosses an anchor boundary, and all
  // float4 plane loads are 16B-aligned: five global_load_b128 per thread.
  float loss = 0.0f;
  int c4 = blockIdx.x * blockDim.x + tid;
  if (c4 < CHUNKS) {
    int c = c4 * 4;
    int a = c / NHW;
    int rem = c - a * NHW;  // multiple of 4
    const float* sbase = out + ((size_t)(b * NA + a) * (5 + NC)) * NHW + rem;
    const v4f* base = (const v4f*)sbase;
    v4f xr4 = base[0 * NHW4];
    v4f yr4 = base[1 * NHW4];
    v4f wr4 = base[2 * NHW4];
    v4f hr4 = base[3 * NHW4];
    v4f cr4 = base[4 * NHW4];

    float aw = d_anchors[2 * a], ah = d_anchors[2 * a + 1];

    float xs[4], ys[4], cf[4], wr[4], hr[4];
    float pw[4], ph[4], parea[4];
    float pex0[4], pex1[4], pey0[4], pey1[4];
    int ji[4], ii[4];
#pragma unroll
    for (int e = 0; e < 4; e++) {
      int rem_e = rem + e;
      int j = rem_e / NW;
      int i = rem_e - j * NW;
      ji[e] = j;
      ii[e] = i;
      xs[e] = fsigmoid(xr4[e]);
      ys[e] = fsigmoid(yr4[e]);
      cf[e] = fsigmoid(cr4[e]);
      wr[e] = wr4[e];
      hr[e] = hr4[e];
      float px = xs[e] + (float)i;
      float py = ys[e] + (float)j;
      pw[e] = __expf(wr4[e]) * aw;
      ph[e] = __expf(hr4[e]) * ah;
      pex0[e] = px - pw[e] * 0.5f;
      pex1[e] = px + pw[e] * 0.5f;
      pey0[e] = py - ph[e] * 0.5f;
      pey1[e] = py + ph[e] * 0.5f;
      parea[e] = pw[e] * ph[e];
    }

    // silent (no-object) test: any valid GT with IoU > 0.6 (division-free).
    // One LDS read of the GT box serves 4 cells.
    bool silent[4] = {false, false, false, false};
    for (int t = 0; t < MAXT; t++) {
      if (!s_valid[t]) break;  // cumulative validity -> early out
      float ex0 = s_ex0[t], ex1 = s_ex1[t];
      float ey0 = s_ey0[t], ey1 = s_ey1[t];
      float gww = s_gw[t], ghh = s_gh[t], ga = s_area[t];
#pragma unroll
      for (int e = 0; e < 4; e++) {
        float mx = fminf(pex0[e], ex0);
        float Mx = fmaxf(pex1[e], ex1);
        float my = fminf(pey0[e], ey0);
        float My = fmaxf(pey1[e], ey1);
        float cw = pw[e] + gww - (Mx - mx);
        float ch = ph[e] + ghh - (My - my);
        if (cw > 0.0f && ch > 0.0f) {
          float inter = cw * ch;
          float uni = parea[e] + ga - inter;
          if (inter > SIL * uni) silent[e] = true;
        }
      }
    }

#pragma unroll
    for (int e = 0; e < 4; e++) {
      int asn = -1;  // at most one winner maps to this (a,j,i)
      for (int t = 0; t < MAXT; t++)
        if (s_win[t] && s_bn[t] == a && s_gj[t] == ji[e] && s_gi[t] == ii[e])
          asn = t;

      float cm = silent[e] ? 0.0f : 1.0f;  // sqrt(1)=1, sqrt(0)=0
      float txv = 0.5f, tyv = 0.5f, twv = 0.0f, thv = 0.0f, tconf = 0.0f;
      if (asn >= 0) {
        cm = 2.23606797749978969f;  // sqrt(OBJECT_SCALE=5)
        txv = s_txv[asn]; tyv = s_tyv[asn];
        twv = s_twv[asn]; thv = s_thv[asn];
        // tconf = IoU(gt[asn], this cell's predicted box)
        float mx = fminf(pex0[e], s_ex0[asn]);
        float Mx = fmaxf(pex1[e], s_ex1[asn]);
        float my = fminf(pey0[e], s_ey0[asn]);
        float My = fmaxf(pey1[e], s_ey1[asn]);
        float cw = pw[e] + s_gw[asn] - (Mx - mx);
        float ch = ph[e] + s_gh[asn] - (My - my);
        float inter = (cw > 0.0f && ch > 0.0f) ? cw * ch : 0.0f;
        float uni = parea[e] + s_area[asn] - inter;
        tconf = inter / fmaxf(uni, 1e-12f);
        // class NLL via log-softmax over 80 logits (rare: <=50 cells/batch)
        const float* lg = sbase + e + 5 * NHW;
        float m = -1e30f;
        for (int cc = 0; cc < NC; cc++) m = fmaxf(m, lg[cc * NHW]);
        float ssum = 0.0f, picked = 0.0f;
        int tc = s_cls[asn];
        for (int cc = 0; cc < NC; cc++) {
          float v = lg[cc * NHW];
          ssum += __expf(v - m);
          if (cc == tc) picked = v;
        }
        loss += -(picked - (m + __logf(ssum)));
      }
      float dx = xs[e] - txv, dy = ys[e] - tyv;
      float dw = wr[e] - twv, dh = hr[e] - thv;
      float dc = cm * cf[e] - cm * tconf;
      loss += 0.5f * (dx * dx + dy * dy + dw * dw + dh * dh + dc * dc);
    }
  }

  // ---- Reduction: wave32 shuffle reduce, then WMMA matrix-unit reduce ----
  for (int off = 16; off > 0; off >>= 1) loss += __shfl_xor(loss, off, 32);
  if ((tid & 31) == 0) s_wsum[tid >> 5] = loss;
  __syncthreads();
  if (tid < 32) {  // wave 0 only -> EXEC all-ones (WMMA requirement)
    float p = (tid < 8) ? s_wsum[tid] : 0.0f;
#if __has_builtin(__builtin_amdgcn_wmma_f32_16x16x4_f32)
    // A = ones(16x4), B holds the 8 partials in unique slots (zeros elsewhere).
    // Every row of D equals the column sums of B, and D-VGPR0 across the 32
    // lanes covers rows M=0 and M=8 => sum(D[0] over lanes) == 2 * sum(B).
    v2f amat = {1.0f, 1.0f};
    v2f bmat = {p, 0.0f};
    v8f cacc = {};
    v8f dres = __builtin_amdgcn_wmma_f32_16x16x4_f32(
        false, amat, false, bmat, (short)0, cacc, false, false);
    float s = dres[0];
    for (int off = 16; off > 0; off >>= 1) s += __shfl_xor(s, off, 32);
    s *= 0.5f;
#else
    float s = p;
    for (int off = 16; off > 0; off >>= 1) s += __shfl_xor(s, off, 32);
#endif
    if (tid == 0) atomicAdd(loss_out, s);
  }
}

extern "C" void kernel_launch(void* const* d_in, const int* in_sizes, int n_in,
                              void* d_out, int out_size, void* d_ws,
                              size_t ws_size, hipStream_t stream) {
  const float* out_t = (const float*)d_in[0];  // (32, 425, 38, 38) f32
  const float* tgt = (const float*)d_in[1];    // (32, 250) f32
  // d_in[2] (features) is unused by the reference loss.
  float* loss = (float*)d_out;
  zero_kernel<<<1, 64, 0, stream>>>(loss, out_size);
  dim3 grid((CHUNKS + 255) / 256, NB);
  region_loss_kernel<<<grid, 256, 0, stream>>>(out_t, tgt, loss);
}